// EinsteinPINN_35596688949622
// MI455X (gfx1250) — compile-verified
//
#include <hip/hip_runtime.h>

// CDNA5 (gfx1250) wave32 WMMA types
typedef __attribute__((ext_vector_type(2))) float v2f;
typedef __attribute__((ext_vector_type(8))) float v8f;

#define HD 256          // hidden size (reference H)
#define KCHUNKS (HD/4)  // K steps of 4 for V_WMMA_F32_16X16X4_F32
#define SPB 64          // samples per block == threads per block (2 waves)

// Branchless tanh: prefer the CDNA5 hardware V_TANH_F32; otherwise compose from
// exp2/rcp TRANS ops (no EXEC-divergent ocml range branches in the WMMA loop).
__device__ __forceinline__ float fast_tanh(float x) {
#if __has_builtin(__builtin_amdgcn_tanhf)
  return __builtin_amdgcn_tanhf(x);
#else
  // tanh(x) = 1 - 2/(e^{2x}+1); exp2 under/overflow gives exact +-1 limits.
  const float z = __builtin_amdgcn_exp2f(x * 2.88539008177792681472f); // 2*log2(e)
  return 1.0f - 2.0f * __builtin_amdgcn_rcpf(z + 1.0f);
#endif
}

__device__ __forceinline__ void inv4x4(const float* m, float* inv) {
  float a00=m[0],a01=m[1],a02=m[2],a03=m[3];
  float a10=m[4],a11=m[5],a12=m[6],a13=m[7];
  float a20=m[8],a21=m[9],a22=m[10],a23=m[11];
  float a30=m[12],a31=m[13],a32=m[14],a33=m[15];
  float b00=a00*a11-a01*a10, b01=a00*a12-a02*a10, b02=a00*a13-a03*a10;
  float b03=a01*a12-a02*a11, b04=a01*a13-a03*a11, b05=a02*a13-a03*a12;
  float b06=a20*a31-a21*a30, b07=a20*a32-a22*a30, b08=a20*a33-a23*a30;
  float b09=a21*a32-a22*a31, b10=a21*a33-a23*a31, b11=a22*a33-a23*a32;
  float det=b00*b11-b01*b10+b02*b09+b03*b08-b04*b07+b05*b06;
  float id=1.0f/det;
  inv[0] =( a11*b11-a12*b10+a13*b09)*id;
  inv[1] =(-a01*b11+a02*b10-a03*b09)*id;
  inv[2] =( a31*b05-a32*b04+a33*b03)*id;
  inv[3] =(-a21*b05+a22*b04-a23*b03)*id;
  inv[4] =(-a10*b11+a12*b08-a13*b07)*id;
  inv[5] =( a00*b11-a02*b08+a03*b07)*id;
  inv[6] =(-a30*b05+a32*b02-a33*b01)*id;
  inv[7] =( a20*b05-a22*b02+a23*b01)*id;
  inv[8] =( a10*b10-a11*b08+a13*b06)*id;
  inv[9] =(-a00*b10+a01*b08-a03*b06)*id;
  inv[10]=( a30*b04-a31*b02+a33*b00)*id;
  inv[11]=(-a20*b04+a21*b02-a23*b00)*id;
  inv[12]=(-a10*b09+a11*b07-a12*b06)*id;
  inv[13]=( a00*b09-a01*b07+a02*b06)*id;
  inv[14]=(-a30*b03+a31*b01-a32*b00)*id;
  inv[15]=( a20*b03-a21*b01+a22*b00)*id;
}

__global__ __launch_bounds__(SPB) void einstein_pinn_kernel(
    const float* __restrict__ x,   // [B,4]
    const float* __restrict__ W1,  // [HD,4]
    const float* __restrict__ b1,  // [HD]
    const float* __restrict__ W2,  // [16,HD]
    const float* __restrict__ b2,  // [16]
    float* __restrict__ ricci_out, // [B,4,4]
    float* __restrict__ riem_out,  // [B,4,4,4,4]
    int Bn)
{
  __shared__ float sW1[HD*4];
  __shared__ float sb1[HD];
  __shared__ float sW2[16*HD];
  __shared__ float sb2[16];
  __shared__ float smlp[SPB][257]; // per-sample 15 vectors x 16 rows; reused as riemann stage
  __shared__ float sdw[SPB][257];  // per-sample dw[p][m][j][k] (256)
  __shared__ float sric[SPB][17];  // per-sample ricci stage

  const int tid = threadIdx.x;
  for (int i = tid; i < HD*4;  i += SPB) sW1[i] = W1[i];
  for (int i = tid; i < HD;    i += SPB) sb1[i] = b1[i];
  for (int i = tid; i < 16*HD; i += SPB) sW2[i] = W2[i];
  if (tid < 16) sb2[tid] = b2[tid];
  __syncthreads();

  const int lane = tid & 31;
  const int wave = tid >> 5;
  const int idx  = lane & 15;   // = M (W2 row) for A, = N (sample) for B/D
  const int half = lane >> 4;   // K/M-half selector per ISA f32 WMMA layouts
  const int blockBase = blockIdx.x * SPB;

  // ---------------- Phase 1: 15 batched W2 matvecs via V_WMMA_F32_16X16X4_F32 ----
  // D[row, sample] = sum_k W2[row,k] * vec_t[sample,k]
  // vec_0 = tanh(u); vec_{1+m} = s*W1[:,m]; vec_{5+pair(m,n)} = -2*t*s*W1[:,m]*W1[:,n]
  for (int g = 0; g < 2; ++g) {
    const int ls0 = wave*32 + g*16 + idx;        // local sample id
    int samp = blockBase + ls0;
    if (samp >= Bn) samp = Bn - 1;               // clamp: keeps EXEC all-ones for WMMA
    const float xs0 = x[samp*4+0], xs1 = x[samp*4+1];
    const float xs2 = x[samp*4+2], xs3 = x[samp*4+3];

    v8f acc[15] = {};

#pragma unroll 2
    for (int kc = 0; kc < KCHUNKS; ++kc) {
      const int ka = kc*4 + 2*half;              // this lane's two K indices
      const int kb = ka + 1;
      const float wa0 = sW1[ka*4+0], wa1 = sW1[ka*4+1], wa2 = sW1[ka*4+2], wa3 = sW1[ka*4+3];
      const float wb0 = sW1[kb*4+0], wb1 = sW1[kb*4+1], wb2 = sW1[kb*4+2], wb3 = sW1[kb*4+3];
      const float ua = wa0*xs0 + wa1*xs1 + wa2*xs2 + wa3*xs3 + sb1[ka];
      const float ub = wb0*xs0 + wb1*xs1 + wb2*xs2 + wb3*xs3 + sb1[kb];
      const float ta = fast_tanh(ua), tb = fast_tanh(ub);
      const float sa = 1.0f - ta*ta, sbv = 1.0f - tb*tb;
      const float pa = -2.0f*ta*sa,  pb = -2.0f*tb*sbv;

      const float wav[4] = {wa0,wa1,wa2,wa3};
      const float wbv[4] = {wb0,wb1,wb2,wb3};
      float va[15], vb[15];
      va[0] = ta; vb[0] = tb;
#pragma unroll
      for (int m = 0; m < 4; ++m) { va[1+m] = sa*wav[m]; vb[1+m] = sbv*wbv[m]; }
      {
        int t = 5;
#pragma unroll
        for (int m = 0; m < 4; ++m)
#pragma unroll
          for (int n = m; n < 4; ++n) {
            va[t] = pa*wav[m]*wav[n];
            vb[t] = pb*wbv[m]*wbv[n];
            ++t;
          }
      }

      v2f A; A.x = sW2[idx*HD + ka]; A.y = sW2[idx*HD + kb]; // A[M=idx, K]
#pragma unroll
      for (int t = 0; t < 15; ++t) {
        v2f Bv; Bv.x = va[t]; Bv.y = vb[t];                  // B[K, N=idx]
        acc[t] = __builtin_amdgcn_wmma_f32_16x16x4_f32(
            false, A, false, Bv, (short)0, acc[t], false, false);
      }
    }

    // D layout: VGPR r, lane -> row = r + 8*half, sample = idx
#pragma unroll
    for (int t = 0; t < 15; ++t)
#pragma unroll
      for (int r = 0; r < 8; ++r)
        smlp[ls0][t*16 + r + 8*half] = acc[t][r];
  }
  __syncthreads();

  // ---------------- Phase 2: per-sample tensor algebra (1 lane = 1 sample) -------
  {
    const int ls = tid;
    const float* Y = &smlp[ls][0];
    const float eta[4] = {-1.f, 1.f, 1.f, 1.f};
    const int pidx[4][4] = {{0,1,2,3},{1,4,5,6},{2,5,7,8},{3,6,8,9}};

    float e[16], de[4][16];
#pragma unroll
    for (int i = 0; i < 4; ++i)
#pragma unroll
      for (int j = 0; j < 4; ++j)
        e[i*4+j] = ((i==j) ? 1.f : 0.f) + 0.1f*(Y[i*4+j] + sb2[i*4+j]);
#pragma unroll
    for (int m = 0; m < 4; ++m)
#pragma unroll
      for (int q = 0; q < 16; ++q)
        de[m][q] = 0.1f * Y[(1+m)*16 + q];

    float f[16];
    inv4x4(e, f);
    float einv[16];                       // e_inv[m,i] = eta_i * f[i,m]
#pragma unroll
    for (int m = 0; m < 4; ++m)
#pragma unroll
      for (int i = 0; i < 4; ++i)
        einv[m*4+i] = eta[i] * f[i*4+m];

    // pw[m,i,j] = sum_n e_inv[n,i] * de[m][n,j]
    float pw[64];
#pragma unroll
    for (int m = 0; m < 4; ++m)
#pragma unroll
      for (int i = 0; i < 4; ++i)
#pragma unroll
        for (int j = 0; j < 4; ++j) {
          float s = 0.f;
#pragma unroll
          for (int n = 0; n < 4; ++n) s += einv[n*4+i] * de[m][n*4+j];
          pw[m*16+i*4+j] = s;
        }

    // r[a,b,c] = (pr[a,b,c] - pr[b,a,c]) / 2   (symmetrization collapses exactly)
    float rr[64];
    {
      float pr[64];
#pragma unroll
      for (int k = 0; k < 4; ++k)
#pragma unroll
        for (int i = 0; i < 4; ++i)
#pragma unroll
          for (int j = 0; j < 4; ++j) {
            float s = 0.f;
#pragma unroll
            for (int m = 0; m < 4; ++m) s += pw[m*16+i*4+j] * einv[m*4+k];
            pr[k*16+i*4+j] = s;
          }
#pragma unroll
      for (int a = 0; a < 4; ++a)
#pragma unroll
        for (int b = 0; b < 4; ++b)
#pragma unroll
          for (int c = 0; c < 4; ++c)
            rr[a*16+b*4+c] = 0.5f*(pr[a*16+b*4+c] - pr[b*16+a*4+c]);
    }

    // dw per derivative direction p -> sdw[ls][p*64 + m*16 + j*4 + k]
    for (int p = 0; p < 4; ++p) {
      float tmp[16], dfp[16];
#pragma unroll
      for (int i = 0; i < 4; ++i)
#pragma unroll
        for (int j = 0; j < 4; ++j) {
          float s = 0.f;
#pragma unroll
          for (int c = 0; c < 4; ++c) s += de[p][i*4+c] * f[c*4+j];
          tmp[i*4+j] = s;
        }
#pragma unroll
      for (int i = 0; i < 4; ++i)
#pragma unroll
        for (int j = 0; j < 4; ++j) {
          float s = 0.f;
#pragma unroll
          for (int c = 0; c < 4; ++c) s += f[i*4+c] * tmp[c*4+j];
          dfp[i*4+j] = -s;                    // d(e^-1) = -e^-1 de e^-1
        }
      float dinv[16];                         // d(e_inv)[m,i] = eta_i * dfp[i,m]
#pragma unroll
      for (int m = 0; m < 4; ++m)
#pragma unroll
        for (int i = 0; i < 4; ++i)
          dinv[m*4+i] = eta[i] * dfp[i*4+m];

      float dpw[64];
#pragma unroll
      for (int m = 0; m < 4; ++m) {
        const float* dd = &smlp[ls][(5 + pidx[p][m])*16];   // dde[p][m] (pre-0.1)
#pragma unroll
        for (int i = 0; i < 4; ++i)
#pragma unroll
          for (int j = 0; j < 4; ++j) {
            float s = 0.f;
#pragma unroll
            for (int n = 0; n < 4; ++n)
              s += dinv[n*4+i]*de[m][n*4+j] + einv[n*4+i]*(0.1f*dd[n*4+j]);
            dpw[m*16+i*4+j] = s;
          }
      }

      float dpr[64];
#pragma unroll
      for (int k = 0; k < 4; ++k)
#pragma unroll
        for (int i = 0; i < 4; ++i)
#pragma unroll
          for (int j = 0; j < 4; ++j) {
            float s = 0.f;
#pragma unroll
            for (int m = 0; m < 4; ++m)
              s += dpw[m*16+i*4+j]*einv[m*4+k] + pw[m*16+i*4+j]*dinv[m*4+k];
            dpr[k*16+i*4+j] = s;
          }

      float drr[64];
#pragma unroll
      for (int a = 0; a < 4; ++a)
#pragma unroll
        for (int b = 0; b < 4; ++b)
#pragma unroll
          for (int c = 0; c < 4; ++c)
            drr[a*16+b*4+c] = 0.5f*(dpr[a*16+b*4+c] - dpr[b*16+a*4+c]);

#pragma unroll
      for (int m = 0; m < 4; ++m)
#pragma unroll
        for (int j = 0; j < 4; ++j)
#pragma unroll
          for (int k = 0; k < 4; ++k) {
            float s = 0.f;
#pragma unroll
            for (int i = 0; i < 4; ++i)
              s += de[p][m*4+i]*eta[i]*rr[i*16+j*4+k]
                 + e[m*4+i]*eta[i]*drr[i*16+j*4+k];
            sdw[ls][p*64 + m*16 + j*4 + k] = s;
          }
    }

    __syncthreads();  // all lanes done reading smlp(dde) -> safe to reuse as stage

    // ricci: temp[m,i] = sum_{n,j} (dw[m][n,i,j]-dw[n][m,i,j]) * e_inv0[n,j]
    float ric1[16];
#pragma unroll
    for (int m = 0; m < 4; ++m)
#pragma unroll
      for (int i = 0; i < 4; ++i) {
        float s = 0.f;
#pragma unroll
        for (int n = 0; n < 4; ++n)
#pragma unroll
          for (int j = 0; j < 4; ++j) {
            float rp = sdw[ls][m*64+n*16+i*4+j] - sdw[ls][n*64+m*16+i*4+j];
            s += rp * f[j*4+n];            // e_inv0[n,j] = f[j,n]
          }
        ric1[m*4+i] = s;
      }
#pragma unroll
    for (int i = 0; i < 4; ++i)
#pragma unroll
      for (int j = 0; j < 4; ++j) {
        float s = 0.f;
#pragma unroll
        for (int m = 0; m < 4; ++m) s += ric1[m*4+j]*einv[m*4+i];
        sric[ls][i*4+j] = s;
      }

    // riemann[k,l,i,j] = (E^T H E)[k,l] per (i,j), H[m,n]=dw[m][n,i,j]-dw[n][m,i,j]
#pragma unroll
    for (int i = 0; i < 4; ++i)
#pragma unroll
      for (int j = 0; j < 4; ++j) {
        float Hm[16];
#pragma unroll
        for (int m = 0; m < 4; ++m)
#pragma unroll
          for (int n = 0; n < 4; ++n)
            Hm[m*4+n] = sdw[ls][m*64+n*16+i*4+j] - sdw[ls][n*64+m*16+i*4+j];
        float T[16];
#pragma unroll
        for (int m = 0; m < 4; ++m)
#pragma unroll
          for (int l = 0; l < 4; ++l) {
            float s = 0.f;
#pragma unroll
            for (int n = 0; n < 4; ++n) s += Hm[m*4+n]*einv[n*4+l];
            T[m*4+l] = s;
          }
#pragma unroll
        for (int k = 0; k < 4; ++k)
#pragma unroll
          for (int l = 0; l < 4; ++l) {
            float s = 0.f;
#pragma unroll
            for (int m = 0; m < 4; ++m) s += einv[m*4+k]*T[m*4+l];
            smlp[ls][k*64 + l*16 + i*4 + j] = s;   // stage for coalesced writeback
          }
      }
  }

  __syncthreads();

  // ---------------- Phase 3: coalesced global writeback --------------------------
  for (int off = tid; off < SPB*256; off += SPB) {
    int s = off >> 8, pos = off & 255;
    int gs = blockBase + s;
    if (gs < Bn) riem_out[(size_t)gs*256 + pos] = smlp[s][pos];
  }
  for (int off = tid; off < SPB*16; off += SPB) {
    int s = off >> 4, pos = off & 15;
    int gs = blockBase + s;
    if (gs < Bn) ricci_out[(size_t)gs*16 + pos] = sric[s][pos];
  }
}

extern "C" void kernel_launch(void* const* d_in, const int* in_sizes, int n_in,
                              void* d_out, int out_size, void* d_ws, size_t ws_size,
                              hipStream_t stream) {
  (void)n_in; (void)out_size; (void)d_ws; (void)ws_size;
  const float* x  = (const float*)d_in[0];
  const float* W1 = (const float*)d_in[1];
  const float* b1 = (const float*)d_in[2];
  const float* W2 = (const float*)d_in[3];
  const float* b2 = (const float*)d_in[4];
  const int Bn = in_sizes[0] / 4;

  float* out   = (float*)d_out;
  float* ricci = out;                       // [B,4,4] first (return order)
  float* riem  = out + (size_t)Bn * 16;     // [B,4,4,4,4] second

  dim3 grid((Bn + SPB - 1) / SPB), block(SPB);
  einstein_pinn_kernel<<<grid, block, 0, stream>>>(x, W1, b1, W2, b2, ricci, riem, Bn);
}